// TransformerBlockWithCache_13709535609486
// MI455X (gfx1250) — compile-verified
//
#include <hip/hip_runtime.h>
#include <math.h>

typedef float v2f __attribute__((ext_vector_type(2)));
typedef float v4f __attribute__((ext_vector_type(4)));
typedef float v8f __attribute__((ext_vector_type(8)));

#define EMBED 2048
#define HEADS 16
#define KV_LEN 4096
#define HEAD_DIM 128

// ---------------------------------------------------------------------------
// zero a region of workspace (accumulators + mask) — must run every call
// ---------------------------------------------------------------------------
__global__ void zero_kernel(float* __restrict__ p, int n) {
  int i = blockIdx.x * blockDim.x + threadIdx.x;
  if (i < n) p[i] = 0.f;
}

// ---------------------------------------------------------------------------
// GEMV  y[j] += sum_i x[i] * W[i][j]   (+ bias once), via V_WMMA_F32_16X16X4_F32
// A (16x4): all 16 rows = x[i..i+3]  -> every D row equals the result
// B (4x16): W[i..i+3][j0..j0+15], streamed with NT loads (read-once, keep
// out of L2 so the attention K/V working set stays resident).
// grid = (32, 8): 4 waves/block * 16 cols each = 2048 cols; K split 8 ways,
// partials combined with atomicAdd (y pre-zeroed).
// ---------------------------------------------------------------------------
__global__ __launch_bounds__(128) void gemv_wmma_kernel(
    const float* __restrict__ x, const float* __restrict__ W,
    const float* __restrict__ bias, float* __restrict__ y, int reluX) {
  const int lane = threadIdx.x & 31;
  const int wave = threadIdx.x >> 5;
  const int tile = blockIdx.x * 4 + wave;       // 0..127
  const int j0   = tile * 16;
  const int i0   = blockIdx.y * 256;            // K-chunk of 256 rows
  const bool hi  = lane >= 16;
  const int  ln  = lane & 15;
  const int  rsel = hi ? 2 : 0;                 // VGPR0 holds K=0/K=2, VGPR1 K=1/K=3

  v8f acc = {0.f, 0.f, 0.f, 0.f, 0.f, 0.f, 0.f, 0.f};

  for (int i = i0; i < i0 + 256; i += 4) {
    float xa = x[i + rsel];
    float xb = x[i + rsel + 1];
    if (reluX) { xa = fmaxf(xa, 0.f); xb = fmaxf(xb, 0.f); }
    v2f a; a.x = xa; a.y = xb;                  // A[m][k] = x[i+k] for all m
    const float* wp = W + (size_t)(i + rsel) * EMBED + j0 + ln;
    v2f b;
    b.x = __builtin_nontemporal_load(wp);        // B[k][n] = W[i+k][j0+n]
    b.y = __builtin_nontemporal_load(wp + EMBED);
    acc = __builtin_amdgcn_wmma_f32_16x16x4_f32(
        false, a, false, b, (short)0, acc, false, false);
  }

  // D row 0 lives in acc[0], lanes 0..15 (N = lane)
  if (!hi) {
    float r = acc[0];
    if (blockIdx.y == 0) r += bias[j0 + ln];
    atomicAdd(&y[j0 + ln], r);
  }
}

// ---------------------------------------------------------------------------
// Roll cache: out[t] = cache[t+1] (t<4095), out[4095] = newrow.
// float4 vectorized; cache read with NT loads (single use); stores keep RT
// temporal hint so the rolled K/V stays in the 192MB L2 for attention.
// Also computes mask[t] = any(v[t]!=0) via race-free same-value stores.
// grid*block == 4096*512 float4 elements exactly.
// ---------------------------------------------------------------------------
__global__ void roll_kernel(const float* __restrict__ cache,
                            const float* __restrict__ nrow,
                            v4f* __restrict__ out,
                            float* __restrict__ maskf) {
  int idx = blockIdx.x * blockDim.x + threadIdx.x;  // float4 index
  int t = idx >> 9;                                 // / 512
  int e = (idx & 511) << 2;                         // float offset in row
  v4f val;
  if (t < KV_LEN - 1)
    val = __builtin_nontemporal_load(
        (const v4f*)(cache + (size_t)(t + 1) * EMBED + e));
  else
    val = *(const v4f*)(nrow + e);
  out[idx] = val;
  if (maskf && (val.x != 0.f || val.y != 0.f || val.z != 0.f || val.w != 0.f))
    maskf[t] = 1.f;
}

// ---------------------------------------------------------------------------
// Attention pass 1 (flash-style partials). grid = (32 chunks, 16 heads),
// block = 256 = 8 waves; each wave owns one timestep at a time (coalesced
// 512B K/V row reads from L2-resident rolled cache), online softmax per
// wave, block combine via LDS.
// Partial record per (h,chunk): [m, s, acc[128]] = 130 floats.
// ---------------------------------------------------------------------------
__global__ __launch_bounds__(256) void attn_partial_kernel(
    const float* __restrict__ q, const float* __restrict__ kc,
    const float* __restrict__ vc, const float* __restrict__ maskf,
    float* __restrict__ part) {
  const int chunk = blockIdx.x;       // 0..31
  const int h     = blockIdx.y;       // 0..15
  const int lane  = threadIdx.x & 31;
  const int wave  = threadIdx.x >> 5; // 0..7
  const int t0    = chunk * 128;

  const v4f qv = *(const v4f*)(q + h * HEAD_DIM + lane * 4);

  float m = -1e30f, s = 0.f;
  v4f acc = {0.f, 0.f, 0.f, 0.f};

  for (int it = 0; it < 16; ++it) {
    int t = t0 + wave + 8 * it;
    const v4f kv = *(const v4f*)(kc + (size_t)t * EMBED + h * HEAD_DIM + lane * 4);
    float d = qv.x * kv.x + qv.y * kv.y + qv.z * kv.z + qv.w * kv.w;
    d += __shfl_xor(d, 16, 32);
    d += __shfl_xor(d, 8, 32);
    d += __shfl_xor(d, 4, 32);
    d += __shfl_xor(d, 2, 32);
    d += __shfl_xor(d, 1, 32);
    float logit = (maskf[t] != 0.f) ? d * 0.08838834764831845f : -1e30f; // 1/sqrt(128)
    float mnew  = fmaxf(m, logit);
    float scale = __expf(m - mnew);
    float p     = (logit > -1e29f) ? __expf(logit - mnew) : 0.f;
    const v4f vv = *(const v4f*)(vc + (size_t)t * EMBED + h * HEAD_DIM + lane * 4);
    s = s * scale + p;
    acc.x = acc.x * scale + p * vv.x;
    acc.y = acc.y * scale + p * vv.y;
    acc.z = acc.z * scale + p * vv.z;
    acc.w = acc.w * scale + p * vv.w;
    m = mnew;
  }

  __shared__ float sm[8], ss[8], sacc[8][HEAD_DIM];
  if (lane == 0) { sm[wave] = m; ss[wave] = s; }
  sacc[wave][lane * 4 + 0] = acc.x;
  sacc[wave][lane * 4 + 1] = acc.y;
  sacc[wave][lane * 4 + 2] = acc.z;
  sacc[wave][lane * 4 + 3] = acc.w;
  __syncthreads();

  int tid = threadIdx.x;
  if (tid < HEAD_DIM) {
    float M = sm[0];
    for (int w = 1; w < 8; ++w) M = fmaxf(M, sm[w]);
    float S = 0.f, A = 0.f;
    for (int w = 0; w < 8; ++w) {
      float e = __expf(sm[w] - M);
      S += ss[w] * e;
      A += e * sacc[w][tid];
    }
    float* pb = part + (size_t)(h * 32 + chunk) * 130;
    if (tid == 0) { pb[0] = M; pb[1] = S; }
    pb[2 + tid] = A;
  }
}

// ---------------------------------------------------------------------------
// Attention pass 2: merge 32 chunk-partials per head, normalize.
// ---------------------------------------------------------------------------
__global__ __launch_bounds__(128) void attn_combine_kernel(
    const float* __restrict__ part, float* __restrict__ values) {
  int h = blockIdx.x;
  int d = threadIdx.x;
  float M = -1e30f;
  for (int c = 0; c < 32; ++c)
    M = fmaxf(M, part[(size_t)(h * 32 + c) * 130]);
  float S = 0.f, A = 0.f;
  for (int c = 0; c < 32; ++c) {
    const float* pb = part + (size_t)(h * 32 + c) * 130;
    float e = __expf(pb[0] - M);
    S += pb[1] * e;
    A += e * pb[2 + d];
  }
  values[h * HEAD_DIM + d] = A / S;
}

// ---------------------------------------------------------------------------
// out = LayerNorm(a + b) * scale + bias   (2048 elems, one block of 256)
// ---------------------------------------------------------------------------
__global__ __launch_bounds__(256) void add_ln_kernel(
    const float* __restrict__ a, const float* __restrict__ b,
    const float* __restrict__ sc, const float* __restrict__ bi,
    float* __restrict__ out) {
  __shared__ float r1[256], r2[256];
  int tid = threadIdx.x;
  float v[8];
  float s = 0.f, s2 = 0.f;
  for (int j = 0; j < 8; ++j) {
    int idx = tid * 8 + j;
    float t = a[idx] + b[idx];
    v[j] = t; s += t; s2 += t * t;
  }
  r1[tid] = s; r2[tid] = s2;
  __syncthreads();
  for (int off = 128; off > 0; off >>= 1) {
    if (tid < off) { r1[tid] += r1[tid + off]; r2[tid] += r2[tid + off]; }
    __syncthreads();
  }
  float mean = r1[0] * (1.f / 2048.f);
  float var  = r2[0] * (1.f / 2048.f) - mean * mean;
  float inv  = 1.f / sqrtf(var + 1e-6f);
  for (int j = 0; j < 8; ++j) {
    int idx = tid * 8 + j;
    out[idx] = (v[j] - mean) * inv * sc[idx] + bi[idx];
  }
}

// ---------------------------------------------------------------------------
extern "C" void kernel_launch(void* const* d_in, const int* in_sizes, int n_in,
                              void* d_out, int out_size, void* d_ws, size_t ws_size,
                              hipStream_t stream) {
  const float* x      = (const float*)d_in[0];
  const float* cache1 = (const float*)d_in[1];   // v cache
  const float* cache2 = (const float*)d_in[2];   // k cache
  const float* Wv = (const float*)d_in[3];  const float* bv = (const float*)d_in[4];
  const float* Wq = (const float*)d_in[5];  const float* bq = (const float*)d_in[6];
  const float* Wk = (const float*)d_in[7];  const float* bk = (const float*)d_in[8];
  const float* Wo = (const float*)d_in[9];  const float* bo = (const float*)d_in[10];
  const float* W1 = (const float*)d_in[11]; const float* b1 = (const float*)d_in[12];
  const float* W2 = (const float*)d_in[13]; const float* b2 = (const float*)d_in[14];
  const float* ln1s = (const float*)d_in[15]; const float* ln1b = (const float*)d_in[16];
  const float* ln2s = (const float*)d_in[17]; const float* ln2b = (const float*)d_in[18];

  float* ws     = (float*)d_ws;
  float* q_v    = ws + 0;       // atomic-accumulated -> zeroed
  float* ki_v   = ws + 2048;    // "
  float* vi_v   = ws + 4096;    // "
  float* oattn  = ws + 6144;    // "
  float* h1v    = ws + 8192;    // "
  float* h2v    = ws + 10240;   // "
  float* maskf  = ws + 12288;   // mask[4096] -> zeroed
  float* values = ws + 16384;   // fully written
  float* x1v    = ws + 18432;   // fully written
  float* part   = ws + 20480;   // 16*32*130 floats, fully written

  float* out_x2 = (float*)d_out;
  float* out_v  = out_x2 + EMBED;
  float* out_k  = out_v + (size_t)KV_LEN * EMBED;

  // 1) zero accumulators + mask
  zero_kernel<<<64, 256, 0, stream>>>(ws, 16384);

  // 2) QKV projections (WMMA fp32 GEMVs, NT weight streams)
  dim3 gg(32, 8);
  gemv_wmma_kernel<<<gg, 128, 0, stream>>>(x, Wq, bq, q_v, 0);
  gemv_wmma_kernel<<<gg, 128, 0, stream>>>(x, Wk, bk, ki_v, 0);
  gemv_wmma_kernel<<<gg, 128, 0, stream>>>(x, Wv, bv, vi_v, 0);

  // 3) roll caches into d_out (v also produces the validity mask)
  roll_kernel<<<8192, 256, 0, stream>>>(cache1, vi_v, (v4f*)out_v, maskf);
  roll_kernel<<<8192, 256, 0, stream>>>(cache2, ki_v, (v4f*)out_k, nullptr);

  // 4) attention over rolled (L2-resident) K/V
  attn_partial_kernel<<<dim3(32, 16), 256, 0, stream>>>(q_v, out_k, out_v, maskf, part);
  attn_combine_kernel<<<16, 128, 0, stream>>>(part, values);

  // 5) output projection, residual+LN, FFN, residual+LN
  gemv_wmma_kernel<<<gg, 128, 0, stream>>>(values, Wo, bo, oattn, 0);
  add_ln_kernel<<<1, 256, 0, stream>>>(x, oattn, ln1s, ln1b, x1v);
  gemv_wmma_kernel<<<gg, 128, 0, stream>>>(x1v, W1, b1, h1v, 0);
  gemv_wmma_kernel<<<gg, 128, 0, stream>>>(h1v, W2, b2, h2v, 1);  // relu on x-operand
  add_ln_kernel<<<1, 256, 0, stream>>>(x1v, h2v, ln2s, ln2b, out_x2);
}